// MultiViewBEVFusionUnified_51436528337641
// MI455X (gfx1250) — compile-verified
//
#include <hip/hip_runtime.h>

typedef float v2f __attribute__((ext_vector_type(2)));
typedef float v8f __attribute__((ext_vector_type(8)));

#define B_    2
#define NV_   4
#define C_    64
#define HF_   128
#define WF_   232
#define NX_   128
#define NY_   128
#define NZ_   8
#define HW_   (HF_ * WF_)          /* 29696  */
#define CHW_  (C_ * HW_)           /* 1900544 */
#define NVOX_ (NX_ * NY_ * NZ_)    /* 131072 */
#define EPS_  1e-5f

// ---------------------------------------------------------------------------
// Kernel 1: stage feats [B,NV,C,H,W] -> [B,NV,H,W,C] (channel-last) so the
// per-pixel 16-channel gathers in the WMMA kernel hit 1-2 cache lines instead
// of 16. Coalesced on both sides via a 64x64 LDS tile. The source is read
// exactly once -> non-temporal loads; the destination is re-read ~17x by the
// gather kernel -> regular (RT) stores so it stays hot in the 192MB L2.
// ---------------------------------------------------------------------------
__global__ __launch_bounds__(256) void tr_kernel(const float* __restrict__ in,
                                                 float* __restrict__ out) {
  __shared__ float tile[C_][C_ + 1];
  const unsigned bv = blockIdx.x / (HW_ / 64);
  const unsigned t0 = (blockIdx.x % (HW_ / 64)) * 64;
  const float* ib = in + (size_t)bv * CHW_;
  float* ob = out + (size_t)bv * CHW_;
  const unsigned tx = threadIdx.x;
#pragma unroll
  for (int r = 0; r < 16; ++r) {
    unsigned c = (tx >> 6) + r * 4;   // 0..63
    unsigned p = tx & 63;             // contiguous across lanes -> coalesced
    tile[c][p] = __builtin_nontemporal_load(&ib[c * HW_ + t0 + p]);
  }
  __syncthreads();
#pragma unroll
  for (int r = 0; r < 16; ++r) {
    unsigned p = (tx >> 6) + r * 4;
    unsigned c = tx & 63;             // contiguous across lanes -> coalesced
    ob[(size_t)(t0 + p) * C_ + c] = tile[c][p];
  }
}

// ---------------------------------------------------------------------------
// Kernel 2: one wave32 per tile of 16 output-order voxels. For each view:
// project, build bilinear weights (OOB masks + valid folded in), then for
// each of the 16 voxels (K-chunk of 4 taps) run 4x V_WMMA_F32_16X16X4_F32
// accumulating D[64 channels x 16 voxels] in 4 C/D fragments.
//
// A (16x4 f32) lane layout: lane L holds (K0,K1) = feat@(y0,xb),(y0,xb+1)
// for channel m*16+L; lane L+16 holds (K2,K3) = same at row y1.
// B (4x16 f32): v0 = {K0 | K2}, v1 = {K1 | K3}; for chunk i only column i is
// nonzero and lane i (/ i+16) already owns those weights -> no shuffle.
// The (y*WF+xb) offset of voxel i is broadcast with one ds_bpermute.
// CS/PS (channel / pixel stride) are compile-time so offset math is
// shifts + 32-bit zext offsets (SADDR-form global loads).
// ---------------------------------------------------------------------------
template <int CS, int PS>
__global__ __launch_bounds__(256) void bev_kernel(const float* __restrict__ tf,
                                                  const float* __restrict__ l2c,
                                                  const float* __restrict__ kmat,
                                                  float* __restrict__ out) {
  const unsigned lane = threadIdx.x & 31;
  const unsigned wid  = threadIdx.x >> 5;
  const unsigned tileId = blockIdx.x * 8 + wid;   // 16384 tiles total
  const unsigned b  = tileId >> 13;               // 8192 tiles per batch
  const unsigned o0 = (tileId & 8191) << 4;       // output-order voxel base
  const unsigned iv = lane & 15;                  // voxel-in-tile & chan row
  const bool lo = lane < 16;

  // voxel world position (o = x*1024 + y*8 + z)
  const unsigned o = o0 + iv;
  const float xw = -25.6f + 0.4f * ((float)(o >> 10) + 0.5f);
  const float yw = -25.6f + 0.4f * ((float)((o >> 3) & 127) + 0.5f);
  const float zw = -1.0f + 0.5f * ((float)(o & 7) + 0.5f);

  v8f acc[4] = {};                                // 64 channels x 16 voxels
  float cnt = 0.f;

  for (int v = 0; v < NV_; ++v) {
    const float* M = l2c + (size_t)(b * NV_ + v) * 16;  // uniform -> s_loads
    const float* K = kmat + (size_t)(b * NV_ + v) * 9;
    float cx = M[0] * xw + M[1] * yw + M[2] * zw + M[3];
    float cy = M[4] * xw + M[5] * yw + M[6] * zw + M[7];
    float cz = M[8] * xw + M[9] * yw + M[10] * zw + M[11];
    float uu = K[0] * cx + K[1] * cy + K[2] * cz;
    float vv = K[3] * cx + K[4] * cy + K[5] * cz;
    float zc = fmaxf(cz, EPS_);
    float px = uu / zc, py = vv / zc;
    bool vld = (cz > EPS_) && (px >= 0.f) && (px < (float)WF_) &&
               (py >= 0.f) && (py < (float)HF_);
    float fvld = vld ? 1.f : 0.f;
    cnt += fvld;

    float xs = px - 0.5f, ys = py - 0.5f;
    float fx0 = floorf(xs), fy0 = floorf(ys);
    float wx = xs - fx0, wy = ys - fy0;
    int x0 = (int)fx0, y0 = (int)fy0;
    float ix0 = (x0 >= 0 && x0 < WF_) ? 1.f : 0.f;
    float ix1 = (x0 >= -1 && x0 < WF_ - 1) ? 1.f : 0.f;
    float iy0 = (y0 >= 0 && y0 < HF_) ? 1.f : 0.f;
    float iy1 = (y0 >= -1 && y0 < HF_ - 1) ? 1.f : 0.f;
    float w00 = (1.f - wx) * (1.f - wy) * ix0 * iy0 * fvld;  // (y0, x0)
    float w10 = wx * (1.f - wy) * ix1 * iy0 * fvld;          // (y0, x0+1)
    float w01 = (1.f - wx) * wy * ix0 * iy1 * fvld;          // (y1, x0)
    float w11 = wx * wy * ix1 * iy1 * fvld;                  // (y1, x0+1)

    // clamp load base; if clamped, the surviving tap moved slots -> swap
    int xb = min(max(x0, 0), WF_ - 2);
    if (x0 != xb) { float t = w00; w00 = w10; w10 = t;
                    t = w01; w01 = w11; w11 = t; }
    int y0c = min(max(y0, 0), HF_ - 1);
    int y1c = min(max(y0 + 1, 0), HF_ - 1);
    int myoff = (lo ? y0c : y1c) * WF_ + xb;      // row offset for my half

    const float* fbv = tf + (size_t)(b * NV_ + v) * CHW_;
#pragma unroll
    for (int ic = 0; ic < 16; ++ic) {
      // broadcast voxel ic's (row,xb) offset: lanes<16 take its y0 row,
      // lanes>=16 take its y1 row
      unsigned off = (unsigned)__builtin_amdgcn_ds_bpermute(
          (int)(((unsigned)ic | (lane & 16)) << 2), myoff);
      v2f bb;
      bb.x = (iv == (unsigned)ic) ? (lo ? w00 : w01) : 0.f;  // K0 | K2
      bb.y = (iv == (unsigned)ic) ? (lo ? w10 : w11) : 0.f;  // K1 | K3
      unsigned pa = off * (unsigned)PS;
      unsigned pb = (off + 1u) * (unsigned)PS;
#pragma unroll
      for (int m = 0; m < 4; ++m) {
        unsigned cidx = ((unsigned)(m * 16) + iv) * (unsigned)CS;
        v2f aa;
        aa.x = fbv[cidx + pa];
        aa.y = fbv[cidx + pb];
        acc[m] = __builtin_amdgcn_wmma_f32_16x16x4_f32(
            false, aa, false, bb, (short)0, acc[m], false, false);
      }
    }
  }

  // mean over valid views; store 16 contiguous floats per half-wave.
  // Output is write-only (67MB) -> non-temporal so the reused staged
  // feature buffer keeps the L2.
  float rcp = 1.f / fmaxf(cnt, 1.f);
  float* ob = out + (size_t)b * (C_ * NVOX_) + o0 + iv;
  const unsigned hi8 = lo ? 0u : 8u;
#pragma unroll
  for (int m = 0; m < 4; ++m) {
#pragma unroll
    for (int r = 0; r < 8; ++r) {
      unsigned c = (unsigned)(m * 16 + r) + hi8;
      __builtin_nontemporal_store(acc[m][r] * rcp, &ob[(size_t)c * NVOX_]);
    }
  }
}

// ---------------------------------------------------------------------------
extern "C" void kernel_launch(void* const* d_in, const int* in_sizes, int n_in,
                              void* d_out, int out_size, void* d_ws, size_t ws_size,
                              hipStream_t stream) {
  const float* feats = (const float*)d_in[0];
  const float* l2c   = (const float*)d_in[1];
  const float* kmat  = (const float*)d_in[2];
  float* out = (float*)d_out;

  const size_t staged_bytes = (size_t)B_ * NV_ * CHW_ * sizeof(float);
  const int gather_blocks = (B_ * NVOX_ / 16) / 8;  // 16384 waves / 8 per blk

  if (ws_size >= staged_bytes) {
    float* tfe = (float*)d_ws;
    tr_kernel<<<B_ * NV_ * (HW_ / 64), 256, 0, stream>>>(feats, tfe);
    bev_kernel<1, C_><<<gather_blocks, 256, 0, stream>>>(tfe, l2c, kmat, out);
  } else {
    bev_kernel<HW_, 1><<<gather_blocks, 256, 0, stream>>>(feats, l2c, kmat, out);
  }
}